// LGONBPLayer_25494925869589
// MI455X (gfx1250) — compile-verified
//
#include <hip/hip_runtime.h>

typedef __attribute__((ext_vector_type(2))) float v2f;
typedef __attribute__((ext_vector_type(4))) float v4f;
typedef __attribute__((ext_vector_type(8))) float v8f;

namespace {

constexpr int kB      = 32;
constexpr int kH      = 384;
constexpr int kW      = 384;
constexpr int kPix    = kH * kW;           // 147456
constexpr int kSlices = 16;                // workgroups per image (streaming passes)
constexpr int kChunk  = kPix / kSlices;    // 9216 pixels per WG
constexpr int kTPB    = 256;               // 8 waves (wave32)
constexpr int kIters  = kChunk / kTPB;     // 36
constexpr int kHist3  = 3 * 256;           // 768
constexpr int kOutD   = 3 * (256 + 128);   // 1152
// # zero-padded neighbor samples per channel histogram: 6H + 6W - 4
constexpr float kPadZeros = (float)(6 * kH + 6 * kW - 4);  // 4604

__device__ __forceinline__ void rgb_to_hsv(float r, float g, float b,
                                           float& h, float& s, float& v) {
  float maxc = fmaxf(r, fmaxf(g, b));
  float minc = fminf(r, fminf(g, b));
  float rng  = maxc - minc;
  float safe_max = (maxc > 0.0f) ? maxc : 1.0f;
  s = (maxc > 0.0f) ? (rng / safe_max) : 0.0f;
  float safe_rng = (rng > 0.0f) ? rng : 1.0f;
  float hh;
  if (maxc == r)      hh = (g - b) / safe_rng;
  else if (maxc == g) hh = 2.0f + (b - r) / safe_rng;
  else                hh = 4.0f + (r - g) / safe_rng;
  float h6 = hh / 6.0f;
  h = (rng > 0.0f) ? (h6 - floorf(h6)) : 0.0f;   // floored mod 1.0
  v = maxc;
}

__device__ __forceinline__ int bin256(float x) {
  // reference: idx = clip(int32((x / (255 + 1e-7)) * 255), 0, 255)
  // 255.0f + 1e-7f rounds to exactly 255.0f in fp32.
  int i = (int)((x / 255.0f) * 255.0f);
  return i < 0 ? 0 : (i > 255 ? 255 : i);
}

}  // namespace

// Pass 1: per-batch channel sums (for the NLBP mean) + weighted LGOP histograms.
__global__ void __launch_bounds__(kTPB)
k_lgop_hist_and_sums(const float* __restrict__ in,
                     unsigned* __restrict__ g_hist,
                     float* __restrict__ g_sum) {
  __shared__ unsigned lh[kHist3];
  __shared__ float lsum[3];
  const int tid = threadIdx.x;
  const int b   = blockIdx.x / kSlices;
  const int sl  = blockIdx.x % kSlices;
  for (int i = tid; i < kHist3; i += kTPB) lh[i] = 0u;
  if (tid < 3) lsum[tid] = 0.0f;
  __syncthreads();

  float sh = 0.f, ss = 0.f, sv = 0.f;
  unsigned z0 = 0u, z1 = 0u, z2 = 0u;  // register fast path for hot bin 0
  const float* bimg = in + (size_t)b * kPix * 3;
  for (int it = 0; it < kIters; ++it) {
    const int p = sl * kChunk + it * kTPB + tid;   // always < kPix
    const int y = p / kW;
    const int x = p - y * kW;
    const float* px = bimg + (size_t)p * 3;
    const float r = px[0], g = px[1], bl = px[2];
    float h, s, v;
    rgb_to_hsv(r, g, bl, h, s, v);
    // multiplicity: # of 8-neighborhood centers this pixel is visible from
    const int cy = 1 + (y > 0) + (y < kH - 1);
    const int cx = 1 + (x > 0) + (x < kW - 1);
    const unsigned w = (unsigned)(cy * cx - 1);    // 8 interior, 5 edge, 3 corner
    const int bh = bin256(h), bs = bin256(s), bv = bin256(v);
    if (bh == 0) z0 += w; else atomicAdd(&lh[bh], w);
    if (bs == 0) z1 += w; else atomicAdd(&lh[256 + bs], w);
    if (bv == 0) z2 += w; else atomicAdd(&lh[512 + bv], w);
    sh += h; ss += s; sv += v;
  }
  if (z0) atomicAdd(&lh[0], z0);
  if (z1) atomicAdd(&lh[256], z1);
  if (z2) atomicAdd(&lh[512], z2);
  atomicAdd(&lsum[0], sh);
  atomicAdd(&lsum[1], ss);
  atomicAdd(&lsum[2], sv);
  __syncthreads();
  if (tid < 3) atomicAdd(&g_sum[b * 3 + tid], lsum[tid]);
  for (int i = tid; i < kHist3; i += kTPB) {
    const unsigned c = lh[i];
    if (c) atomicAdd(&g_hist[b * kHist3 + i], c);
  }
}

// Pass 2: recompute HSV, count values strictly above the per-batch channel mean.
__global__ void __launch_bounds__(kTPB)
k_nlbp_counts(const float* __restrict__ in,
              const float* __restrict__ g_sum,
              unsigned* __restrict__ g_cnt) {
  __shared__ unsigned lc[3];
  const int tid = threadIdx.x;
  const int b   = blockIdx.x / kSlices;
  const int sl  = blockIdx.x % kSlices;
  if (tid < 3) lc[tid] = 0u;
  __syncthreads();
  const float mh = g_sum[b * 3 + 0] / (float)kPix;
  const float ms = g_sum[b * 3 + 1] / (float)kPix;
  const float mv = g_sum[b * 3 + 2] / (float)kPix;
  unsigned ch = 0u, cs = 0u, cv = 0u;
  const float* bimg = in + (size_t)b * kPix * 3;
  for (int it = 0; it < kIters; ++it) {
    const int p = sl * kChunk + it * kTPB + tid;
    const float* px = bimg + (size_t)p * 3;
    float h, s, v;
    rgb_to_hsv(px[0], px[1], px[2], h, s, v);
    ch += (h > mh) ? 1u : 0u;
    cs += (s > ms) ? 1u : 0u;
    cv += (v > mv) ? 1u : 0u;
  }
  atomicAdd(&lc[0], ch);
  atomicAdd(&lc[1], cs);
  atomicAdd(&lc[2], cv);
  __syncthreads();
  if (tid < 3) {
    const unsigned c = lc[tid];
    if (c) atomicAdd(&g_cnt[b * 3 + tid], c);
  }
}

// Pass 3: assemble the 1152-d descriptor, Sum(x^2) via the diagonal of A*A^T
// using chained v_wmma_f32_16x16x4_f32, then L2-normalize. One wave per batch.
__global__ void __launch_bounds__(32)
k_finalize(const unsigned* __restrict__ g_hist,
           const unsigned* __restrict__ g_cnt,
           float* __restrict__ out) {
  __shared__ float xv[kOutD];
  const int b = blockIdx.x;
  const int l = threadIdx.x;  // 0..31, full wave32, EXEC all ones

  for (int j = l; j < kOutD; j += 32) {
    const int sec = j / 384;          // channel: h, s, v
    const int off = j - sec * 384;
    float val;
    if (off < 256) {                  // LGOP histogram section
      val = (float)g_hist[b * kHist3 + sec * 256 + off];
      if (off == 0) val += kPadZeros; // zero-padded neighbor samples
    } else {                          // NLBP section: only bins 0 and 126 populated
      const int bin = off - 256;
      const unsigned c = g_cnt[b * 3 + sec];
      val = (bin == 0) ? (float)(kPix - (int)c)
          : (bin == 126) ? (float)c : 0.0f;
    }
    xv[j] = val;
  }
  __syncthreads();

  // Sum of squares = trace(A * A^T), A = 16x4 f32 chunks of xv; 18 chunks * 64 = 1152.
  v8f acc = {0.f, 0.f, 0.f, 0.f, 0.f, 0.f, 0.f, 0.f};
  const int half = l >> 4;        // 0: lanes 0-15, 1: lanes 16-31
  const int m    = l & 15;
  for (int t = 0; t < 18; ++t) {
    const v4f q = *(const v4f*)&xv[t * 64 + 4 * m];  // 4 consecutive elems per lane
    v2f a, bb;
    if (half == 0) {              // A lanes 0-15: K=0,1 ; B lanes 0-15: rows K=0,2
      a.x = q.x; a.y = q.y;
      bb.x = q.x; bb.y = q.z;
    } else {                      // A lanes 16-31: K=2,3 ; B lanes 16-31: rows K=1,3
      a.x = q.z; a.y = q.w;
      bb.x = q.y; bb.y = q.w;
    }
    acc = __builtin_amdgcn_wmma_f32_16x16x4_f32(
        /*neg_a=*/false, a, /*neg_b=*/false, bb,
        /*c_mod=*/(short)0, acc, /*reuse_a=*/false, /*reuse_b=*/false);
  }
  // Diagonal D[m,m]: m=0..7 -> lane m, acc[m]; m=8..15 -> lane m+16, acc[m-8].
  float d = 0.0f;
#pragma unroll
  for (int i = 0; i < 8; ++i) {
    const bool pick = (l == i) || (l == 24 + i);
    d = pick ? acc[i] : d;
  }
  for (int o = 16; o > 0; o >>= 1) d += __shfl_xor(d, o, 32);

  const float nrm = rsqrtf(fmaxf(d, 1e-12f));
  for (int j = l; j < kOutD; j += 32) out[b * kOutD + j] = xv[j] * nrm;
}

extern "C" void kernel_launch(void* const* d_in, const int* in_sizes, int n_in,
                              void* d_out, int out_size, void* d_ws, size_t ws_size,
                              hipStream_t stream) {
  (void)in_sizes; (void)n_in; (void)out_size;
  const float* in = (const float*)d_in[0];

  // Workspace layout: [B*768 u32 hists][B*3 f32 sums][B*3 u32 counts]
  const size_t hist_bytes = (size_t)kB * kHist3 * sizeof(unsigned);
  const size_t sum_bytes  = (size_t)kB * 3 * sizeof(float);
  const size_t cnt_bytes  = (size_t)kB * 3 * sizeof(unsigned);
  unsigned* g_hist = (unsigned*)d_ws;
  float*    g_sum  = (float*)((char*)d_ws + hist_bytes);
  unsigned* g_cnt  = (unsigned*)((char*)d_ws + hist_bytes + sum_bytes);
  (void)ws_size;

  hipMemsetAsync(d_ws, 0, hist_bytes + sum_bytes + cnt_bytes, stream);

  k_lgop_hist_and_sums<<<dim3(kB * kSlices), dim3(kTPB), 0, stream>>>(in, g_hist, g_sum);
  k_nlbp_counts<<<dim3(kB * kSlices), dim3(kTPB), 0, stream>>>(in, g_sum, g_cnt);
  k_finalize<<<dim3(kB), dim3(32), 0, stream>>>(g_hist, g_cnt, (float*)d_out);
}